// ConvPoolReadout_85109071938348
// MI455X (gfx1250) — compile-verified
//
#include <hip/hip_runtime.h>
#include <math.h>

#define N_NODES 65536
#define NPER    512
#define NGRAPH  128
#define FDIM    256
#define DEG     16
#define NEDGE   (N_NODES * DEG)
#define KSEL    256

#define KCHUNK       16                       // K rows staged per chunk
#define CHUNK_FLOATS (KCHUNK * FDIM)          // 4096 floats = 16 KB
#define CHUNK_BYTES  (CHUNK_FLOATS * 4)

typedef float v2f __attribute__((ext_vector_type(2)));
typedef float v8f __attribute__((ext_vector_type(8)));

// ---------------------------------------------------------------- utilities
__global__ __launch_bounds__(256)
void zero_int_kernel(int* __restrict__ p, int n) {
    int i = blockIdx.x * blockDim.x + threadIdx.x;
    for (; i < n; i += gridDim.x * blockDim.x) p[i] = 0;
}

__global__ __launch_bounds__(256)
void count_kernel(const int* __restrict__ src, const int* __restrict__ dst,
                  int* __restrict__ cnt, int* __restrict__ odeg) {
    int e = blockIdx.x * blockDim.x + threadIdx.x;
    if (e >= NEDGE) return;
    atomicAdd(&cnt[dst[e]], 1);
    atomicAdd(&odeg[src[e]], 1);
}

__global__ __launch_bounds__(256)
void norm_kernel(const int* __restrict__ cnt, const int* __restrict__ odeg,
                 float* __restrict__ snorm, float* __restrict__ dnorm) {
    int n = blockIdx.x * blockDim.x + threadIdx.x;
    if (n >= N_NODES) return;
    float od = fmaxf((float)odeg[n], 1.0f);
    float id = fmaxf((float)cnt[n], 1.0f);
    snorm[n] = 1.0f / sqrtf(od);
    dnorm[n] = 1.0f / sqrtf(id);
}

// single-block exclusive scan over N_NODES counts -> rowptr[N_NODES+1]
__global__ __launch_bounds__(1024)
void scan_kernel(const int* __restrict__ cnt, int* __restrict__ rowptr) {
    __shared__ int sm[1024];
    __shared__ int carry;
    const int tid = threadIdx.x;
    if (tid == 0) carry = 0;
    __syncthreads();
    for (int base = 0; base < N_NODES; base += 1024) {
        int v = cnt[base + tid];
        sm[tid] = v;
        __syncthreads();
        for (int off = 1; off < 1024; off <<= 1) {
            int t = (tid >= off) ? sm[tid - off] : 0;
            __syncthreads();
            sm[tid] += t;
            __syncthreads();
        }
        int incl = sm[tid];
        int c = carry;
        rowptr[base + tid] = c + incl - v;      // exclusive
        __syncthreads();
        if (tid == 1023) carry = c + incl;      // chunk total
        __syncthreads();
    }
    if (tid == 0) rowptr[N_NODES] = carry;
}

__global__ __launch_bounds__(256)
void fill_kernel(const int* __restrict__ src, const int* __restrict__ dst,
                 const float* __restrict__ ef, const int* __restrict__ rowptr,
                 int* __restrict__ fill, int* __restrict__ colb,
                 float* __restrict__ ewb) {
    int e = blockIdx.x * blockDim.x + threadIdx.x;
    if (e >= NEDGE) return;
    int d = dst[e];
    int slot = rowptr[d] + atomicAdd(&fill[d], 1);
    colb[slot] = src[e];
    ewb[slot]  = ef[e];
}

// ---------------------------------------------------------------- WMMA GEMM
// h = (feat * src_norm) @ W     [65536x256] x [256x256]
// Block: 8 waves = 2 mTiles x 4 nGroups -> 32 rows x 256 cols.
// W staged per 16-row k-chunk into LDS with global_load_async_to_lds_b128
// (double buffered, ASYNCcnt pipelined); B fragments read from LDS.
__global__ __launch_bounds__(256)
void gemm_kernel(const float* __restrict__ feat, const float* __restrict__ W,
                 const float* __restrict__ snorm, float* __restrict__ h) {
    __shared__ float wbuf[2 * CHUNK_FLOATS];   // 32 KB double buffer

    const int tid    = threadIdx.x;
    const int waveI  = tid >> 5;               // 0..7
    const int lane   = tid & 31;
    const int mTile  = (blockIdx.x << 1) | (waveI >> 2);
    const int nGroup = waveI & 3;
    const int m      = lane & 15;
    const int khalf  = (lane >> 4) << 1;       // lanes 16..31 carry K+2,K+3
    const int row    = mTile * 16 + m;
    const int nBase  = nGroup * 64;
    const float sn   = snorm[row];
    const float* __restrict__ arow = feat + (size_t)row * FDIM;

    // stage chunk kc (W rows [kc*16, kc*16+16) = contiguous 16 KB) into buffer.
    // LDS dest address is derived from the real &wbuf[...] pointer (flat addr
    // low 32 bits == DS byte offset) so the array escapes into the asm and the
    // "memory" clobber makes the compiler honor the DMA writes.
    auto stage = [&](int kc, int buf) {
        const float* gp = W + (size_t)kc * CHUNK_FLOATS + tid * 4;
        float*       lp = &wbuf[buf * CHUNK_FLOATS + tid * 4];
        #pragma unroll
        for (int j = 0; j < 4; ++j) {           // 4 x (256 thr x 16 B) = 16 KB
            uint32_t ldsAddr = (uint32_t)(uintptr_t)(lp + j * 1024);
            asm volatile("global_load_async_to_lds_b128 %0, %1, off"
                         :: "v"(ldsAddr), "v"(gp + j * 1024)
                         : "memory");
        }
    };

    v8f c0 = {}, c1 = {}, c2 = {}, c3 = {};
    stage(0, 0);
    for (int kc = 0; kc < FDIM / KCHUNK; ++kc) {
        if (kc + 1 < FDIM / KCHUNK) {
            stage(kc + 1, (kc + 1) & 1);
            asm volatile("s_wait_asynccnt 4" ::: "memory");  // chunk kc landed
        } else {
            asm volatile("s_wait_asynccnt 0" ::: "memory");
        }
        __syncthreads();
        const float* wb = &wbuf[(kc & 1) * CHUNK_FLOATS];
        #pragma unroll
        for (int ks = 0; ks < KCHUNK; ks += 4) {
            const int k0 = kc * KCHUNK + ks;
            v2f a;
            a.x = arow[k0 + khalf]     * sn;
            a.y = arow[k0 + khalf + 1] * sn;
            const float* w0 = wb + (ks + khalf) * FDIM + nBase + m;
            v2f b0, b1, b2, b3;
            b0.x = w0[0];  b0.y = w0[FDIM];
            b1.x = w0[16]; b1.y = w0[FDIM + 16];
            b2.x = w0[32]; b2.y = w0[FDIM + 32];
            b3.x = w0[48]; b3.y = w0[FDIM + 48];
            c0 = __builtin_amdgcn_wmma_f32_16x16x4_f32(false, a, false, b0, (short)0, c0, false, false);
            c1 = __builtin_amdgcn_wmma_f32_16x16x4_f32(false, a, false, b1, (short)0, c1, false, false);
            c2 = __builtin_amdgcn_wmma_f32_16x16x4_f32(false, a, false, b2, (short)0, c2, false, false);
            c3 = __builtin_amdgcn_wmma_f32_16x16x4_f32(false, a, false, b3, (short)0, c3, false, false);
        }
        __syncthreads();   // all waves done reading buf before it is restaged
    }

    // D layout: vgpr j -> row j (lanes 0..15) / row j+8 (lanes 16..31), N = lane&15
    const int rowOff = (lane >> 4) * 8;
    const int ncol   = lane & 15;
    for (int j = 0; j < 8; ++j) {
        float* p = h + (size_t)(mTile * 16 + rowOff + j) * FDIM + nBase + ncol;
        p[0]  = c0[j];
        p[16] = c1[j];
        p[32] = c2[j];
        p[48] = c3[j];
    }
}

// ---------------------------------------------------------------- aggregation
// out[n] = relu(dst_norm[n] * sum_{e: dst=n} h[src_e] * ew_e + b)
__global__ __launch_bounds__(256)
void agg_relu_kernel(const float* __restrict__ h, const int* __restrict__ colb,
                     const float* __restrict__ ewb, const int* __restrict__ rowptr,
                     const float* __restrict__ dnorm, const float* __restrict__ bias,
                     float* __restrict__ out) {
    const int node = blockIdx.x * 8 + (threadIdx.x >> 5);
    const int lane = threadIdx.x & 31;
    const int f0   = lane * 8;
    float4 a0 = {0, 0, 0, 0}, a1 = {0, 0, 0, 0};
    const int s0 = rowptr[node], s1 = rowptr[node + 1];
    for (int s = s0; s < s1; ++s) {
        const int   src = colb[s];
        const float w   = ewb[s];
        if (s + 1 < s1)
            __builtin_prefetch(h + (size_t)colb[s + 1] * FDIM + f0, 0, 0);
        const float4* hp = (const float4*)(h + (size_t)src * FDIM + f0);
        float4 v0 = hp[0], v1 = hp[1];
        a0.x += v0.x * w; a0.y += v0.y * w; a0.z += v0.z * w; a0.w += v0.w * w;
        a1.x += v1.x * w; a1.y += v1.y * w; a1.z += v1.z * w; a1.w += v1.w * w;
    }
    const float dn = dnorm[node];
    const float4* bp = (const float4*)(bias + f0);
    float4 b0 = bp[0], b1 = bp[1];
    float4 o0, o1;
    o0.x = fmaxf(a0.x * dn + b0.x, 0.f); o0.y = fmaxf(a0.y * dn + b0.y, 0.f);
    o0.z = fmaxf(a0.z * dn + b0.z, 0.f); o0.w = fmaxf(a0.w * dn + b0.w, 0.f);
    o1.x = fmaxf(a1.x * dn + b1.x, 0.f); o1.y = fmaxf(a1.y * dn + b1.y, 0.f);
    o1.z = fmaxf(a1.z * dn + b1.z, 0.f); o1.w = fmaxf(a1.w * dn + b1.w, 0.f);
    float4* op = (float4*)(out + (size_t)node * FDIM + f0);
    op[0] = o0; op[1] = o1;
}

// score[n] = sum_f | out[n,f] - dn * sum_{e: dst=n} out[src_e,f]*snorm[src_e] |
__global__ __launch_bounds__(256)
void score_kernel(const float* __restrict__ out, const int* __restrict__ colb,
                  const int* __restrict__ rowptr, const float* __restrict__ snorm,
                  const float* __restrict__ dnorm, float* __restrict__ score) {
    const int node = blockIdx.x * 8 + (threadIdx.x >> 5);
    const int lane = threadIdx.x & 31;
    const int f0   = lane * 8;
    float4 a0 = {0, 0, 0, 0}, a1 = {0, 0, 0, 0};
    const int s0 = rowptr[node], s1 = rowptr[node + 1];
    for (int s = s0; s < s1; ++s) {
        const int   src = colb[s];
        const float w   = snorm[src];
        if (s + 1 < s1)
            __builtin_prefetch(out + (size_t)colb[s + 1] * FDIM + f0, 0, 0);
        const float4* op = (const float4*)(out + (size_t)src * FDIM + f0);
        float4 v0 = op[0], v1 = op[1];
        a0.x += v0.x * w; a0.y += v0.y * w; a0.z += v0.z * w; a0.w += v0.w * w;
        a1.x += v1.x * w; a1.y += v1.y * w; a1.z += v1.z * w; a1.w += v1.w * w;
    }
    const float dn = dnorm[node];
    const float4* sp = (const float4*)(out + (size_t)node * FDIM + f0);
    float4 u0 = sp[0], u1 = sp[1];
    float v = fabsf(u0.x - dn * a0.x) + fabsf(u0.y - dn * a0.y) +
              fabsf(u0.z - dn * a0.z) + fabsf(u0.w - dn * a0.w) +
              fabsf(u1.x - dn * a1.x) + fabsf(u1.y - dn * a1.y) +
              fabsf(u1.z - dn * a1.z) + fabsf(u1.w - dn * a1.w);
    for (int off = 16; off > 0; off >>= 1) v += __shfl_down(v, off, 32);
    if (lane == 0) score[node] = v;
}

// ---------------------------------------------------------------- top-k + readout
// rank selection; exactly reproduces stable argsort(-score)[:, :K]
__global__ __launch_bounds__(512)
void topk_kernel(const float* __restrict__ score, int* __restrict__ perm) {
    __shared__ float sc[NPER];
    const int g = blockIdx.x, tid = threadIdx.x;
    sc[tid] = score[g * NPER + tid];
    __syncthreads();
    const float my = sc[tid];
    int rank = 0;
    for (int j = 0; j < NPER; ++j) {
        float o = sc[j];
        rank += (o > my) || ((o == my) && (j < tid));
    }
    if (rank < KSEL) perm[g * KSEL + rank] = g * NPER + tid;
}

__global__ __launch_bounds__(256)
void pool_readout_kernel(const float* __restrict__ out, const int* __restrict__ perm,
                         float* __restrict__ pooled, float* __restrict__ readout) {
    __shared__ int pm[KSEL];
    const int g = blockIdx.x, f = threadIdx.x;
    pm[f] = perm[g * KSEL + f];
    __syncthreads();
    float sum = 0.f, mx = -INFINITY;
    for (int r = 0; r < KSEL; ++r) {
        float v = out[(size_t)pm[r] * FDIM + f];
        pooled[((size_t)g * KSEL + r) * FDIM + f] = v;
        sum += v;
        mx = fmaxf(mx, v);
    }
    readout[(size_t)g * 2 * FDIM + f]        = sum * (1.0f / KSEL);
    readout[(size_t)g * 2 * FDIM + FDIM + f] = mx;
}

// ---------------------------------------------------------------- launcher
extern "C" void kernel_launch(void* const* d_in, const int* in_sizes, int n_in,
                              void* d_out, int out_size, void* d_ws, size_t ws_size,
                              hipStream_t stream) {
    const float* feat  = (const float*)d_in[0];
    const float* efeat = (const float*)d_in[1];
    const float* W     = (const float*)d_in[2];
    const float* bias  = (const float*)d_in[3];
    const int*   src   = (const int*)d_in[4];
    const int*   dst   = (const int*)d_in[5];

    char* ws = (char*)d_ws;
    float* h      = (float*)(ws);                       // 64 MB
    float* out    = (float*)(ws + 67108864ull);         // 64 MB
    int*   colb   = (int*)  (ws + 134217728ull);        // 4 MB
    float* ewb    = (float*)(ws + 138412032ull);        // 4 MB
    int*   cnt    = (int*)  (ws + 142606336ull);        // 256 KB (in-degree)
    int*   odeg   = (int*)  (ws + 142868480ull);        // 256 KB
    int*   fill   = (int*)  (ws + 143130624ull);        // 256 KB
    int*   rowptr = (int*)  (ws + 143392768ull);        // 256 KB + 4
    float* snorm  = (float*)(ws + 143655168ull);
    float* dnorm  = (float*)(ws + 143917312ull);
    float* score  = (float*)(ws + 144179456ull);
    int*   perm   = (int*)  (ws + 144441600ull);        // 128 KB

    float* pooled  = (float*)d_out;
    float* readout = (float*)d_out + (size_t)NGRAPH * KSEL * FDIM;

    // cnt, odeg, fill are contiguous: zero all three in one launch
    zero_int_kernel<<<768, 256, 0, stream>>>(cnt, 3 * N_NODES);
    count_kernel<<<NEDGE / 256, 256, 0, stream>>>(src, dst, cnt, odeg);
    norm_kernel<<<N_NODES / 256, 256, 0, stream>>>(cnt, odeg, snorm, dnorm);
    scan_kernel<<<1, 1024, 0, stream>>>(cnt, rowptr);
    fill_kernel<<<NEDGE / 256, 256, 0, stream>>>(src, dst, efeat, rowptr, fill, colb, ewb);
    gemm_kernel<<<N_NODES / 32, 256, 0, stream>>>(feat, W, snorm, h);
    agg_relu_kernel<<<N_NODES / 8, 256, 0, stream>>>(h, colb, ewb, rowptr, dnorm, bias, out);
    score_kernel<<<N_NODES / 8, 256, 0, stream>>>(out, colb, rowptr, snorm, dnorm, score);
    topk_kernel<<<NGRAPH, 512, 0, stream>>>(score, perm);
    pool_readout_kernel<<<NGRAPH, 256, 0, stream>>>(out, perm, pooled, readout);
}